// RNN_51092930953625
// MI455X (gfx1250) — compile-verified
//
#include <hip/hip_runtime.h>

// RNN scan: h_t = tanh(W_ih*x_t + W_hh*h_{t-1} + b), out_t = W_fc*h_t + b_fc
// SEQ=1048576, HID=10, IN=OUT=1.  Latency-bound sequential scan -> single
// persistent wave32; 10x10 matvec per step done as 3 chained
// V_WMMA_F32_16X16X4_F32 rank-4 updates with h broadcast into the A layout
// via ds_bpermute.  FC head: per-step ds_store of h into an LDS row,
// per-32-step block each lane reduces its own row (off the recurrence path).

typedef __attribute__((ext_vector_type(2))) float v2f;
typedef __attribute__((ext_vector_type(4))) float v4f;
typedef __attribute__((ext_vector_type(8))) float v8f;

#define SEQ_N 1048576
#define FEAT_N 3
#define HID_N 10
#define ROWF 36              // LDS row pitch in floats (144 B, 16-aligned, 2-way bank conflict max)

#if __has_builtin(__builtin_amdgcn_tanhf)
#define TANHF(x) __builtin_amdgcn_tanhf(x)
#else
#define TANHF(x) tanhf(x)
#endif

__device__ __forceinline__ float bperm_f(int idx_bytes, float v) {
    return __int_as_float(__builtin_amdgcn_ds_bpermute(idx_bytes, __float_as_int(v)));
}

__global__ __launch_bounds__(32, 1)
void rnn_scan_wmma(const float* __restrict__ src,   // (SEQ,1,3), use [:,:,0]
                   const float* __restrict__ W_ih,  // (10,1)
                   const float* __restrict__ W_hh,  // (10,10)
                   const float* __restrict__ b_ih,  // (10)
                   const float* __restrict__ b_hh,  // (10)
                   const float* __restrict__ W_fc,  // (1,10)
                   const float* __restrict__ b_fc,  // (1)
                   float* __restrict__ out)         // (SEQ,1,1)
{
    __shared__ __align__(16) float hbuf[32 * ROWF];   // row k = h after step k

    const int lane = threadIdx.x;          // 0..31, single wave
    const int n    = lane & 15;            // WMMA column index (N)
    const int hi2  = (lane >> 4) << 1;     // 0 for lanes 0-15, 2 for lanes 16-31

    // ---- B operand: W_hh^T zero-padded to 12x16, resident in registers. ----
    // B[k][n] = W_hh[n][k].  V_WMMA_F32_16X16X4_F32 B layout (4x16, 2 VGPRs):
    // VGPR0: rows K=0 (lanes 0-15) / K=2 (lanes 16-31); VGPR1: K=1 / K=3.
    v2f Bm[3];
#pragma unroll
    for (int c = 0; c < 3; ++c) {
        const int k0 = 4 * c + hi2;
        const int k1 = k0 + 1;
        Bm[c].x = (k0 < HID_N && n < HID_N) ? W_hh[n * HID_N + k0] : 0.0f;
        Bm[c].y = (k1 < HID_N && n < HID_N) ? W_hh[n * HID_N + k1] : 0.0f;
    }

    // Constant bpermute byte-indices to build A (h broadcast by K) from
    // lane-indexed vh.
    int idxA[3][2];
#pragma unroll
    for (int c = 0; c < 3; ++c) {
        idxA[c][0] = (4 * c + hi2) << 2;
        idxA[c][1] = (4 * c + 1 + hi2) << 2;
    }

    const float wih  = (n < HID_N) ? W_ih[n] : 0.0f;             // mirrored both halves
    const float bsum = (n < HID_N) ? (b_ih[n] + b_hh[n]) : 0.0f; // mirrored both halves
    const float bfc  = b_fc[0];

    // FC weights as uniform scalars (SGPRs) for the block-end row reduce.
    float sfc[HID_N];
#pragma unroll
    for (int j = 0; j < HID_N; ++j) sfc[j] = W_fc[j];

    float vh = 0.0f;          // lane k holds h_k (k<16; mirrored in 16-31)
    v8f acc = {};             // WMMA C/D carry; only element 0 is meaningful

    const int NBLK = SEQ_N / 32;

    // Double-buffered, coalesced x loads: one global_load_b32 per 32 steps.
    float xv = src[(size_t)lane * FEAT_N];

    float* myrow = &hbuf[lane];          // per-step store target (col = lane)
    const v4f* rrow = (const v4f*)&hbuf[lane * ROWF];  // block-end gather row

    for (int blk = 0; blk < NBLK; ++blk) {
        const int t0 = blk * 32;
        float xv_next = xv;
        if (blk + 1 < NBLK)
            xv_next = src[(size_t)(t0 + 32 + lane) * FEAT_N];

#pragma unroll
        for (int k = 0; k < 32; ++k) {
            // x_t: uniform scalar via v_readlane from the block's x register
            const float x_t =
                __int_as_float(__builtin_amdgcn_readlane(__float_as_int(xv), k));

            // C row 0 = b + W_ih * x_t  (per column n); rows 1-7 are don't-care
            acc[0] = fmaf(x_t, wih, bsum);

            // Build A chunks from vh (6 independent bpermutes)
            v2f a0, a1, a2;
            a0.x = bperm_f(idxA[0][0], vh);
            a0.y = bperm_f(idxA[0][1], vh);
            a1.x = bperm_f(idxA[1][0], vh);
            a1.y = bperm_f(idxA[1][1], vh);
            a2.x = bperm_f(idxA[2][0], vh);
            a2.y = bperm_f(idxA[2][1], vh);

            // D = A*B + C accumulated over K = 0..11 (rows 10,11 zero-padded)
            v8f d = __builtin_amdgcn_wmma_f32_16x16x4_f32(
                false, a0, false, Bm[0], (short)0, acc, false, false);
            d = __builtin_amdgcn_wmma_f32_16x16x4_f32(
                false, a1, false, Bm[1], (short)0, d, false, false);
            d = __builtin_amdgcn_wmma_f32_16x16x4_f32(
                false, a2, false, Bm[2], (short)0, d, false, false);

            // h_new[n] sits in D element 0, lane n (columns >= 10 are exactly 0)
            vh  = TANHF(d[0]);
            acc = d;   // accumulator registers carried in place

            // Dump h_t row for the FC head: 1 ds_store_b32 at immediate offset
            myrow[k * ROWF] = vh;
        }

        // ---- FC head for this block: lane s reduces LDS row s ----
        // (single wave: DS ops are in-order, compiler inserts s_wait_dscnt)
        const v4f h03 = rrow[0];             // h[0..3]
        const v4f h47 = rrow[1];             // h[4..7]
        const float h8 = hbuf[lane * ROWF + 8];
        const float h9 = hbuf[lane * ROWF + 9];
        float o = bfc;
        o = fmaf(h03.x, sfc[0], o);
        o = fmaf(h03.y, sfc[1], o);
        o = fmaf(h03.z, sfc[2], o);
        o = fmaf(h03.w, sfc[3], o);
        o = fmaf(h47.x, sfc[4], o);
        o = fmaf(h47.y, sfc[5], o);
        o = fmaf(h47.z, sfc[6], o);
        o = fmaf(h47.w, sfc[7], o);
        o = fmaf(h8,    sfc[8], o);
        o = fmaf(h9,    sfc[9], o);

        // Coalesced store of 32 outputs
        out[t0 + lane] = o;
        xv = xv_next;
    }
}

extern "C" void kernel_launch(void* const* d_in, const int* in_sizes, int n_in,
                              void* d_out, int out_size, void* d_ws, size_t ws_size,
                              hipStream_t stream) {
    const float* src  = (const float*)d_in[0];
    const float* W_ih = (const float*)d_in[1];
    const float* W_hh = (const float*)d_in[2];
    const float* b_ih = (const float*)d_in[3];
    const float* b_hh = (const float*)d_in[4];
    const float* W_fc = (const float*)d_in[5];
    const float* b_fc = (const float*)d_in[6];
    float* out = (float*)d_out;

    // Sequential scan: exactly one wave32 does everything in registers.
    rnn_scan_wmma<<<1, 32, 0, stream>>>(src, W_ih, W_hh, b_ih, b_hh, W_fc, b_fc, out);
}